// GraphUNet2_867583393847
// MI455X (gfx1250) — compile-verified
//
#include <hip/hip_runtime.h>
#include <math.h>

// Graph U-Net forward for MI455X (gfx1250, wave32, WMMA).
// B=256 graphs, N=1024 nodes, E=8192 edges, fp32 end-to-end.
// GEMM layers use V_WMMA_F32_16X16X4_F32 (exact fp32 matrix pipe),
// with compile-time K/N, b64 A-fragment loads, and LDS-staged transposed W.

#define BB 256
#define NN 1024
#define EE 8192

typedef __attribute__((ext_vector_type(2))) float v2f;
typedef __attribute__((ext_vector_type(8))) float v8f;

// ---------------- elementwise copy (float4 / b128) ----------------
__global__ void k_copy4(const float4* __restrict__ in, float4* __restrict__ out, long n4) {
  long i = (long)blockIdx.x * blockDim.x + threadIdx.x;
  if (i < n4) out[i] = in[i];
}

__global__ void k_add(const float* __restrict__ a, const float* __restrict__ b,
                      float* __restrict__ o, long n) {
  long i = (long)blockIdx.x * blockDim.x + threadIdx.x;
  if (i < n) o[i] = a[i] + b[i];
}

// ---------------- GIN scatter-add: out[dst] += H[src] ----------------
// H is the *unmodified* feature array; out was pre-initialized to a copy of H.
__global__ void k_scatter(const float* __restrict__ H, const int* __restrict__ src,
                          const int* __restrict__ dst, float* __restrict__ out,
                          int Nc, int F) {
  long i = (long)blockIdx.x * blockDim.x + threadIdx.x;
  if (i >= (long)BB * EE) return;
  int b = (int)(i / EE), e = (int)(i % EE);
  int s = src[(long)b * EE + e];
  if (s < 0) return;                       // masked edge
  int d = dst[(long)b * EE + e];
  const float* sr = H + ((long)b * Nc + s) * F;
  float* dr = out + ((long)b * Nc + d) * F;
  for (int f = 0; f < F; ++f) unsafeAtomicAdd(&dr[f], sr[f]);
}

// ---------------- fused GEMM + bias + BN + ReLU via fp32 WMMA ----------------
// out[b, m, n] = relu( (sum_k A[b,m,k]*W[k,n] + bias[n] - mu[n]) * rsqrt(var[n]+eps)*g[n] + be[n] )
// One wave computes a 16x16 tile, K stepped by 4 with V_WMMA_F32_16X16X4_F32.
// A frag (16x4 f32, ISA 7.12.2): lane L -> row M=L%16; VGPR0/1 hold K = 2*(L/16)+{0,1}.
// B frag (4x16 f32), mirrored:   lane L -> col N=L%16; VGPR0/1 hold K = 2*(L/16)+{0,1}.
// C/D (16x16 f32): VGPR t: lanes 0-15 -> row m0+t, lanes 16-31 -> row m0+t+8; col = L%16.
// Rows >= M use a clamped (valid) address; their outputs are simply not stored —
// no divergence, no selects in the hot loop (output row m depends only on A row m).
template <int K, int N>
__global__ void __launch_bounds__(32)
k_gemm_bn_relu(const float* __restrict__ A, const float* __restrict__ W,
               const float* __restrict__ bias, const float* __restrict__ g,
               const float* __restrict__ be, const float* __restrict__ mu,
               const float* __restrict__ var, float* __restrict__ out,
               int M, long sA, long sO) {
  constexpr int WROW = K + 4;                       // pad to dodge bank conflicts
  __shared__ __align__(16) float wt[16 * WROW];     // wt[n][k] = W[k][n0+n]
  int b = blockIdx.z;
  const float* Ab = A + (long)b * sA;
  float* Ob = out + (long)b * sO;
  int m0 = blockIdx.x * 16, n0 = blockIdx.y * 16;
  int lane = threadIdx.x;
  int hi = lane >> 4;
  int lp = lane & 15;

  // stage transposed W tile: lanes 0-15 take even k rows, 16-31 odd rows (coalesced)
  for (int kk = hi; kk < K; kk += 2)
    wt[lp * WROW + kk] = W[(long)kk * N + n0 + lp];
  __syncthreads();

  int rowA = m0 + lp;
  int rowc = rowA < M ? rowA : M - 1;               // clamp address, skip store later
  const float* arow = Ab + (long)rowc * K;
  const float* wrow = &wt[lp * WROW];

  v8f acc = {};
#pragma unroll 8
  for (int k0 = 0; k0 < K; k0 += 4) {
    v2f a  = *(const v2f*)(arow + k0 + 2 * hi);     // global_load_b64
    v2f bb = *(const v2f*)(wrow + k0 + 2 * hi);     // ds_load_b64
    acc = __builtin_amdgcn_wmma_f32_16x16x4_f32(false, a, false, bb,
                                                (short)0, acc, false, false);
  }

  int n = n0 + lp;
  float rs = rsqrtf(var[n] + 1e-5f);
  float al = rs * g[n];
  float shift = bias[n] - mu[n];
  float beta = be[n];
#pragma unroll
  for (int t = 0; t < 8; ++t) {
    int row = m0 + t + 8 * hi;
    if (row < M) {
      float o = (acc[t] + shift) * al + beta;
      Ob[(long)row * N + n] = fmaxf(o, 0.f);
    }
  }
}

// ---------------- TopK pooling ----------------
__global__ void k_score(const float* __restrict__ H, const float* __restrict__ w,
                        float* __restrict__ score, int Nc, int F) {
  long i = (long)blockIdx.x * blockDim.x + threadIdx.x;
  if (i >= (long)BB * Nc) return;
  int b = (int)(i / Nc), n = (int)(i % Nc);
  const float* row = H + ((long)b * Nc + n) * F;
  float dot = 0.f, wn = 0.f;
  for (int f = 0; f < F; ++f) { dot += row[f] * w[f]; wn += w[f] * w[f]; }
  score[i] = dot / sqrtf(wn);
}

// rank = exact position in descending sort with index tie-break == jax.lax.top_k perm.
// kept node: new id = rank; xp[rank] = H[n] * tanh(score[n]); mapping[n] = rank or -1.
__global__ void k_select(const float* __restrict__ score, const float* __restrict__ H,
                         float* __restrict__ xp, int* __restrict__ mapping,
                         int Nc, int F, int k) {
  long i = (long)blockIdx.x * blockDim.x + threadIdx.x;
  if (i >= (long)BB * Nc) return;
  int b = (int)(i / Nc), n = (int)(i % Nc);
  const float* sc = score + (long)b * Nc;
  float s = sc[n];
  int rank = 0;
  for (int j = 0; j < Nc; ++j) {
    float sj = sc[j];
    rank += (sj > s) || (sj == s && j < n);
  }
  if (rank < k) {
    mapping[(long)b * Nc + n] = rank;
    float t = tanhf(s);
    const float* row = H + ((long)b * Nc + n) * F;
    float* orow = xp + ((long)b * k + rank) * F;
    for (int f = 0; f < F; ++f) orow[f] = row[f] * t;
  } else {
    mapping[(long)b * Nc + n] = -1;
  }
}

__global__ void k_remap(const int* __restrict__ src, const int* __restrict__ dst,
                        const int* __restrict__ mapping, int Nc,
                        int* __restrict__ ns, int* __restrict__ nd) {
  long i = (long)blockIdx.x * blockDim.x + threadIdx.x;
  if (i >= (long)BB * EE) return;
  int b = (int)(i / EE);
  int s = src[i], d = dst[i];
  int ms = (s >= 0) ? mapping[(long)b * Nc + s] : -1;
  int md = (d >= 0) ? mapping[(long)b * Nc + d] : -1;
  bool ok = (ms >= 0) && (md >= 0);
  ns[i] = ok ? ms : -1;
  nd[i] = ok ? md : -1;
}

// cat([max over nodes, mean over nodes]) -> out[b, 0:F]=max, out[b, F:2F]=mean
__global__ void k_readout(const float* __restrict__ xp, float* __restrict__ out,
                          int k, int F) {
  int i = blockIdx.x * blockDim.x + threadIdx.x;
  if (i >= BB * F) return;
  int b = i / F, f = i % F;
  const float* base = xp + (long)b * k * F + f;
  float mx = -INFINITY, sm = 0.f;
  for (int n = 0; n < k; ++n) { float v = base[(long)n * F]; mx = fmaxf(mx, v); sm += v; }
  out[(long)b * 2 * F + f] = mx;
  out[(long)b * 2 * F + F + f] = sm / (float)k;
}

// final: logits = (d2 + x1) @ W[64,10] + bias; out = log_softmax(logits)
__global__ void k_dec1(const float* __restrict__ d2, const float* __restrict__ x1,
                       const float* __restrict__ W, const float* __restrict__ bias,
                       float* __restrict__ out) {
  int b = blockIdx.x * blockDim.x + threadIdx.x;
  if (b >= BB) return;
  float h[64];
  for (int f = 0; f < 64; ++f) h[f] = d2[b * 64 + f] + x1[b * 64 + f];
  float lg[10]; float mx = -INFINITY;
  for (int c = 0; c < 10; ++c) {
    float s = bias[c];
    for (int f = 0; f < 64; ++f) s += h[f] * W[f * 10 + c];
    lg[c] = s; mx = fmaxf(mx, s);
  }
  float se = 0.f;
  for (int c = 0; c < 10; ++c) se += expf(lg[c] - mx);
  float lse = mx + logf(se);
  for (int c = 0; c < 10; ++c) out[b * 10 + c] = lg[c] - lse;
}

#define PF(i) ((const float*)d_in[(i)])

extern "C" void kernel_launch(void* const* d_in, const int* in_sizes, int n_in,
                              void* d_out, int out_size, void* d_ws, size_t ws_size,
                              hipStream_t stream) {
  (void)in_sizes; (void)n_in; (void)out_size; (void)ws_size;
  const float* x = (const float*)d_in[0];
  const int* es = (const int*)d_in[1];
  const int* ed = (const int*)d_in[2];

  // ---- workspace layout (~208 MB): two big ping-pong regions + edges + smalls ----
  char* ws = (char*)d_ws;
  size_t off = 0;
  auto alloc = [&](size_t bytes) -> void* {
    void* p = ws + off; off += (bytes + 255) & ~(size_t)255; return p;
  };
  const size_t BIG = (size_t)BB * 656 * 128 * sizeof(float); // max stage tenant (86 MB)
  float* bufA = (float*)alloc(BIG);
  float* bufB = (float*)alloc(BIG);
  int* s1  = (int*)alloc((size_t)BB * EE * 4);
  int* d1  = (int*)alloc((size_t)BB * EE * 4);
  int* s2  = (int*)alloc((size_t)BB * EE * 4);
  int* d2i = (int*)alloc((size_t)BB * EE * 4);
  float* score   = (float*)alloc((size_t)BB * NN * 4);
  int*   mapping = (int*)alloc((size_t)BB * NN * 4);
  float* x1 = (float*)alloc(BB * 64 * 4);
  float* x2 = (float*)alloc(BB * 128 * 4);
  float* x3 = (float*)alloc(BB * 256 * 4);
  float* t0 = (float*)alloc(BB * 128 * 4);
  float* t1 = (float*)alloc(BB * 128 * 4);
  float* t2 = (float*)alloc(BB * 128 * 4);
  float* t3 = (float*)alloc(BB * 64 * 4);
  float* t4 = (float*)alloc(BB * 64 * 4);

  auto gl = [](long n) { return dim3((unsigned)((n + 255) / 256)); };
  const long nBE = (long)BB * EE;

  // ================= conv1 (GIN 64 -> 32 -> 32) =================
  long n_h0 = (long)BB * NN * 64;
  k_copy4<<<gl(n_h0 / 4), 256, 0, stream>>>((const float4*)x, (float4*)bufA, n_h0 / 4);
  k_scatter<<<gl(nBE), 256, 0, stream>>>(x, es, ed, bufA, NN, 64);
  k_gemm_bn_relu<64, 32><<<dim3(64, 2, BB), 32, 0, stream>>>(bufA, PF(3), PF(4), PF(5), PF(6),
      PF(7), PF(8), bufB, 1024, (long)1024 * 64, (long)1024 * 32);
  k_gemm_bn_relu<32, 32><<<dim3(64, 2, BB), 32, 0, stream>>>(bufB, PF(9), PF(10), PF(11), PF(12),
      PF(13), PF(14), bufA, 1024, (long)1024 * 32, (long)1024 * 32);

  // ================= pool1 (1024 -> 820) + readout x1 =================
  k_score<<<gl((long)BB * 1024), 256, 0, stream>>>(bufA, PF(15), score, 1024, 32);
  k_select<<<gl((long)BB * 1024), 256, 0, stream>>>(score, bufA, bufB, mapping, 1024, 32, 820);
  k_remap<<<gl(nBE), 256, 0, stream>>>(es, ed, mapping, 1024, s1, d1);
  k_readout<<<gl(BB * 32), 256, 0, stream>>>(bufB, x1, 820, 32);

  // ================= conv2 (GIN 32 -> 64 -> 64) =================
  long n2 = (long)BB * 820 * 32;
  k_copy4<<<gl(n2 / 4), 256, 0, stream>>>((const float4*)bufB, (float4*)bufA, n2 / 4);
  k_scatter<<<gl(nBE), 256, 0, stream>>>(bufB, s1, d1, bufA, 820, 32);
  k_gemm_bn_relu<32, 64><<<dim3(52, 4, BB), 32, 0, stream>>>(bufA, PF(16), PF(17), PF(18), PF(19),
      PF(20), PF(21), bufB, 820, (long)820 * 32, (long)820 * 64);
  k_gemm_bn_relu<64, 64><<<dim3(52, 4, BB), 32, 0, stream>>>(bufB, PF(22), PF(23), PF(24), PF(25),
      PF(26), PF(27), bufA, 820, (long)820 * 64, (long)820 * 64);

  // ================= pool2 (820 -> 656) + readout x2 =================
  k_score<<<gl((long)BB * 820), 256, 0, stream>>>(bufA, PF(28), score, 820, 64);
  k_select<<<gl((long)BB * 820), 256, 0, stream>>>(score, bufA, bufB, mapping, 820, 64, 656);
  k_remap<<<gl(nBE), 256, 0, stream>>>(s1, d1, mapping, 820, s2, d2i);
  k_readout<<<gl(BB * 64), 256, 0, stream>>>(bufB, x2, 656, 64);

  // ================= conv3 (GIN 64 -> 128 -> 128) =================
  long n3 = (long)BB * 656 * 64;
  k_copy4<<<gl(n3 / 4), 256, 0, stream>>>((const float4*)bufB, (float4*)bufA, n3 / 4);
  k_scatter<<<gl(nBE), 256, 0, stream>>>(bufB, s2, d2i, bufA, 656, 64);
  k_gemm_bn_relu<64, 128><<<dim3(41, 8, BB), 32, 0, stream>>>(bufA, PF(29), PF(30), PF(31), PF(32),
      PF(33), PF(34), bufB, 656, (long)656 * 64, (long)656 * 128);
  k_gemm_bn_relu<128, 128><<<dim3(41, 8, BB), 32, 0, stream>>>(bufB, PF(35), PF(36), PF(37), PF(38),
      PF(39), PF(40), bufA, 656, (long)656 * 128, (long)656 * 128);

  // ================= pool3 (656 -> 525) + readout x3 =================
  k_score<<<gl((long)BB * 656), 256, 0, stream>>>(bufA, PF(41), score, 656, 128);
  k_select<<<gl((long)BB * 656), 256, 0, stream>>>(score, bufA, bufB, mapping, 656, 128, 525);
  k_readout<<<gl(BB * 128), 256, 0, stream>>>(bufB, x3, 525, 128);

  // ================= decoder (batch = 1, M = 256 rows) =================
  k_gemm_bn_relu<256, 128><<<dim3(16, 8, 1), 32, 0, stream>>>(x3, PF(42), PF(43), PF(44), PF(45),
      PF(46), PF(47), t0, 256, 0, 0);
  k_gemm_bn_relu<128, 128><<<dim3(16, 8, 1), 32, 0, stream>>>(t0, PF(48), PF(49), PF(50), PF(51),
      PF(52), PF(53), t1, 256, 0, 0);
  k_add<<<gl((long)BB * 128), 256, 0, stream>>>(t1, x2, t2, (long)BB * 128);
  k_gemm_bn_relu<128, 64><<<dim3(16, 4, 1), 32, 0, stream>>>(t2, PF(54), PF(55), PF(56), PF(57),
      PF(58), PF(59), t3, 256, 0, 0);
  k_gemm_bn_relu<64, 64><<<dim3(16, 4, 1), 32, 0, stream>>>(t3, PF(60), PF(61), PF(62), PF(63),
      PF(64), PF(65), t4, 256, 0, 0);
  k_dec1<<<gl(BB), 256, 0, stream>>>(t4, x1, PF(66), PF(67), (float*)d_out);
}